// LSVAE_23751169147394
// MI455X (gfx1250) — compile-verified
//
#include <hip/hip_runtime.h>

// Problem constants (from reference)
#define T_   256
#define B_   128
#define D_   32
#define DU_  8
#define S_   4
#define LDP  33   // LDS row pitch (padding -> conflict-free column access)
#define LOG2PI_F 1.8378770664093453f

typedef __attribute__((ext_vector_type(2))) float v2f;
typedef __attribute__((ext_vector_type(8))) float v8f;

// ---------------------------------------------------------------------------
// Async global->LDS tile machinery (GLOBAL_LOAD_ASYNC_TO_LDS_B128, ASYNCcnt).
// Falls back to synchronous float4 copies if the builtins are unavailable.
// Builtin wants: arg0 = v4i* in AS1 (global), arg1 = v4i* in AS3 (LDS),
// then imm offset, imm cpol. Pointers built via integer round-trip to avoid
// address-space cast legality issues (AS3 pointer value = LDS byte offset).
// ---------------------------------------------------------------------------
#if defined(__has_builtin)
#if __has_builtin(__builtin_amdgcn_global_load_async_to_lds_b128) && \
    __has_builtin(__builtin_amdgcn_s_wait_asynccnt)
#define USE_ASYNC 1
#endif
#endif
#ifndef USE_ASYNC
#define USE_ASYNC 0
#endif

#if USE_ASYNC
typedef __attribute__((ext_vector_type(4))) int v4i;
typedef __attribute__((address_space(1))) v4i* gvp_t;
typedef __attribute__((address_space(3))) v4i* lvp_t;
#define ASYNC_WAIT(n) __builtin_amdgcn_s_wait_asynccnt(n)
#else
#define ASYNC_WAIT(n) ((void)0)
#endif

// Copy one 32x32 f32 tile (4KB, 16B-aligned) global -> LDS; 8 x b128 per lane.
__device__ inline void tile_load_async(const float* __restrict__ g, float* l, int r) {
#if USE_ASYNC
  gvp_t gp = (gvp_t)(unsigned long long)g;            // 64-bit global VA
  lvp_t lp = (lvp_t)(unsigned int)(unsigned long long)l; // low 32 bits = LDS offset
#pragma unroll
  for (int i = 0; i < 8; ++i)
    __builtin_amdgcn_global_load_async_to_lds_b128(gp + (i * 32 + r),
                                                   lp + (i * 32 + r), 0, 0);
#else
  const float4* gs = (const float4*)g;
  float4* ls = (float4*)l;
#pragma unroll
  for (int i = 0; i < 8; ++i) ls[i * 32 + r] = gs[i * 32 + r];
#endif
}

// ---------------------------------------------------------------------------
// Wave-level 32x32x32 fp32 GEMM on LDS tiles using V_WMMA_F32_16X16X4_F32.
// Cd = (NEGOUT? -1:1) * op(Am)*op(Bm) + (ADDC? Cadd : 0)
// Fragment layouts per CDNA5 ISA 7.12.2:
//   A (16x4 f32): lane l -> m = l&15 ; reg v -> k = k0 + v + 2*(l>>4)
//   B (4x16 f32): lane l -> n = l&15 ; reg v -> k = k0 + v + 2*(l>>4)
//   C/D (16x16 f32): lane l -> n = l&15 ; reg q -> m = q + 8*(l>>4)
// ---------------------------------------------------------------------------
template<bool TA, bool TB, bool ADDC, bool NEGOUT>
__device__ inline void gemm32(float* __restrict__ Cd, const float* Am, const float* Bm,
                              const float* Cadd, int lane) {
  const int lm = lane & 15;
  const int hf = lane >> 4;
#pragma unroll
  for (int mi = 0; mi < 32; mi += 16) {
#pragma unroll
    for (int ni = 0; ni < 32; ni += 16) {
      v8f acc;
#pragma unroll
      for (int q = 0; q < 8; ++q)
        acc[q] = ADDC ? Cadd[(mi + q + 8 * hf) * LDP + ni + lm] : 0.0f;
#pragma unroll
      for (int k0 = 0; k0 < 32; k0 += 4) {
        v2f a, b;
        if (TA) { // op(A)[m][k] = Am[k][m]
          a.x = Am[(k0 + 2 * hf) * LDP + mi + lm];
          a.y = Am[(k0 + 2 * hf + 1) * LDP + mi + lm];
        } else {
          a.x = Am[(mi + lm) * LDP + k0 + 2 * hf];
          a.y = Am[(mi + lm) * LDP + k0 + 2 * hf + 1];
        }
        if (TB) { // op(B)[k][n] = Bm[n][k]
          b.x = Bm[(ni + lm) * LDP + k0 + 2 * hf];
          b.y = Bm[(ni + lm) * LDP + k0 + 2 * hf + 1];
        } else {
          b.x = Bm[(k0 + 2 * hf) * LDP + ni + lm];
          b.y = Bm[(k0 + 2 * hf + 1) * LDP + ni + lm];
        }
        acc = __builtin_amdgcn_wmma_f32_16x16x4_f32(false, a, false, b,
                                                    (short)0, acc, false, false);
      }
#pragma unroll
      for (int q = 0; q < 8; ++q)
        Cd[(mi + q + 8 * hf) * LDP + ni + lm] = NEGOUT ? -acc[q] : acc[q];
    }
  }
}

// ---------------------------------------------------------------------------
// Gauss-Jordan inversion of SPD 32x32 (no pivoting). One wave, lane = row.
// M destroyed, X <- M^{-1}. Returns log|det(M)| (same value on all lanes).
// ---------------------------------------------------------------------------
__device__ inline float gj_invert(float* M, float* X, int r) {
  for (int j = 0; j < 32; ++j) X[r * LDP + j] = (r == j) ? 1.0f : 0.0f;
  __syncthreads();
  float ld = 0.0f;
  for (int k = 0; k < 32; ++k) {
    float piv  = M[k * LDP + k];      // broadcast read
    float pinv = 1.0f / piv;
    ld += __logf(piv);                // SPD -> pivots > 0
    float f    = M[r * LDP + k] * pinv;
    bool  isk  = (r == k);
    for (int j = 0; j < 32; ++j) {
      float pr = M[k * LDP + j];      // loads precede stores (LDS in-order)
      float xr = X[k * LDP + j];
      float mv = M[r * LDP + j];
      float xv = X[r * LDP + j];
      M[r * LDP + j] = isk ? pr * pinv : mv - f * pr;
      X[r * LDP + j] = isk ? xr * pinv : xv - f * xr;
    }
    __syncthreads();
  }
  return ld;
}

// ---------------------------------------------------------------------------
// In-place Cholesky (lower) of SPD 32x32, lane = row. Returns logdet(M).
// ---------------------------------------------------------------------------
__device__ inline float chol32(float* M, int r) {
  float ld = 0.0f;
  for (int k = 0; k < 32; ++k) {
    __syncthreads();
    float dk = M[k * LDP + k];
    float lk = __fsqrt_rn(dk);
    float il = 1.0f / lk;
    ld += __logf(lk);
    float a = M[r * LDP + k] * il;
    if (r == k)      M[r * LDP + k] = lk;
    else if (r > k)  M[r * LDP + k] = a;
    __syncthreads();
    if (r > k) {
      float lrk = M[r * LDP + k];
      for (int j = k + 1; j <= r; ++j) M[r * LDP + j] -= lrk * M[j * LDP + k];
    }
  }
  __syncthreads();
  return 2.0f * ld;
}

// Solve L y = v (forward substitution), v in LDS, lane = row.
__device__ inline void trisolve_fwd(const float* L, float* v, int r) {
  for (int k = 0; k < 32; ++k) {
    __syncthreads();
    float yk = v[k] / L[k * LDP + k];
    if (r == k)      v[r] = yk;
    else if (r > k)  v[r] -= L[r * LDP + k] * yk;
  }
  __syncthreads();
}

// Solve L^T x = v (backward substitution).
__device__ inline void trisolve_bwd(const float* L, float* v, int r) {
  for (int k = 31; k >= 0; --k) {
    __syncthreads();
    float xk = v[k] / L[k * LDP + k];
    if (r == k)      v[r] = xk;
    else if (r < k)  v[r] -= L[k * LDP + r] * xk;
  }
  __syncthreads();
}

__device__ inline float wredsum(float v) {
#pragma unroll
  for (int off = 16; off > 0; off >>= 1) v += __shfl_xor(v, off, 32);
  return v;
}

// ---------------------------------------------------------------------------
// Prep: Sw_inv = inv(Sigma_w), AT_Sw = A^T Sw_inv, AT_Sw_A = AT_Sw A,
// logdet(Sw_inv); zero the output accumulator.
// ---------------------------------------------------------------------------
__global__ __launch_bounds__(32) void prep_kernel(
    const float* __restrict__ A, const float* __restrict__ Sigma_w,
    float* __restrict__ swinv_g, float* __restrict__ atsw_g,
    float* __restrict__ atswa_g, float* __restrict__ ldsw_g,
    float* __restrict__ out) {
  __shared__ float sM[32 * LDP], sX[32 * LDP], sA[32 * LDP], sW[32 * LDP];
  const int r = threadIdx.x;
  for (int i = 0; i < 32; ++i) {
    sM[i * LDP + r] = Sigma_w[i * 32 + r];
    sA[i * LDP + r] = A[i * 32 + r];
  }
  __syncthreads();
  float ld = gj_invert(sM, sX, r);            // sX = Sw_inv, ld = logdet(Sigma_w)
  for (int i = 0; i < 32; ++i) swinv_g[i * 32 + r] = sX[i * LDP + r];
  // AT_Sw = A^T * Sw_inv
  gemm32<true, false, false, false>(sW, sA, sX, nullptr, r);
  __syncthreads();
  for (int i = 0; i < 32; ++i) atsw_g[i * 32 + r] = sW[i * LDP + r];
  // AT_Sw_A = AT_Sw * A
  gemm32<false, false, false, false>(sM, sW, sA, nullptr, r);
  __syncthreads();
  for (int i = 0; i < 32; ++i) atswa_g[i * 32 + r] = sM[i * LDP + r];
  if (r == 0) ldsw_g[0] = -ld;                // logdet(Sw_inv)
  // zero KL accumulator (B_=128)
  out[r] = 0.0f; out[32 + r] = 0.0f; out[64 + r] = 0.0f; out[96 + r] = 0.0f;
}

// ---------------------------------------------------------------------------
// Forward information filter. One wave per batch element. Per step:
//   M = conc_p + AT_Sw_A ; G = M^{-1}            (Gauss-Jordan)
//   pred_conc = Sw_inv - H G H^T, H = Sw_inv A   (2 WMMA GEMMs)
//   pred_inf  = H (G inf_p) + pred_conc (B u)    (matvecs)
//   posterior = pred + observation potential
// obs_conc[t+1] tile is async-loaded into a double buffer while step t runs.
// ---------------------------------------------------------------------------
__global__ __launch_bounds__(32) void forward_kernel(
    const float* __restrict__ obs_inf, const float* __restrict__ obs_conc,
    const float* __restrict__ u, const float* __restrict__ Bmat,
    const float* __restrict__ prior_inf, const float* __restrict__ prior_conc,
    const float* __restrict__ swinv_g, const float* __restrict__ atsw_g,
    const float* __restrict__ atswa_g,
    float* __restrict__ post_inf, float* __restrict__ post_conc) {
  __shared__ float sC[32 * LDP], sI[32 * LDP], sT[32 * LDP];
  __shared__ float sSwi[32 * LDP], sATSw[32 * LDP], sATSwA[32 * LDP];
  __shared__ __align__(16) float sObs[2][D_ * D_];   // async double buffer
  __shared__ float sB[D_ * DU_];
  __shared__ float sInf[32], sV1[32], sBu[32];
  const int r = threadIdx.x;
  const int b = blockIdx.x;

  for (int i = 0; i < 32; ++i) {
    sSwi[i * LDP + r]   = swinv_g[i * 32 + r];
    sATSw[i * LDP + r]  = atsw_g[i * 32 + r];
    sATSwA[i * LDP + r] = atswa_g[i * 32 + r];
  }
  for (int k = r; k < D_ * DU_; k += 32) sB[k] = Bmat[k];

  // t = 0 : posterior = obs + prior
  {
    size_t vb = (size_t)b * D_;
    size_t cb = (size_t)b * D_ * D_;
    sInf[r] = obs_inf[vb + r] + prior_inf[r];
    for (int i = 0; i < 32; ++i)
      sC[i * LDP + r] = obs_conc[cb + i * 32 + r] + prior_conc[i * 32 + r];
    __syncthreads();
    post_inf[vb + r] = sInf[r];
    for (int i = 0; i < 32; ++i) post_conc[cb + i * 32 + r] = sC[i * LDP + r];
  }
  // kick off async load of obs_conc tile for t=1
  tile_load_async(obs_conc + ((size_t)(1 * B_ + b)) * D_ * D_, sObs[1], r);

  for (int t = 1; t < T_; ++t) {
    // start async load of next observation tile while this step computes
    if (t + 1 < T_)
      tile_load_async(obs_conc + ((size_t)((t + 1) * B_ + b)) * D_ * D_,
                      sObs[(t + 1) & 1], r);
    // M = conc_p + AT_Sw_A (row-wise, own row only)
    for (int j = 0; j < 32; ++j) sC[r * LDP + j] += sATSwA[r * LDP + j];
    __syncthreads();
    (void)gj_invert(sC, sI, r);               // sI = G = M^{-1}
    // v1 = G * inf_p ; bu = Bmat * u[t-1]
    float v1 = 0.0f;
    for (int j = 0; j < 32; ++j) v1 += sI[r * LDP + j] * sInf[j];
    sV1[r] = v1;
    const float* ut = u + ((size_t)(t - 1) * B_ + b) * DU_;
    float bu = 0.0f;
    for (int c = 0; c < DU_; ++c) bu += sB[r * DU_ + c] * ut[c];
    sBu[r] = bu;
    __syncthreads();
    // sT = -(H * G), H = AT_Sw^T  (Sw_inv symmetric)
    gemm32<true, false, false, true>(sT, sATSw, sI, nullptr, r);
    __syncthreads();
    // pred_conc = Sw_inv + sT * AT_Sw   (= Sw_inv - H G H^T)
    gemm32<false, false, true, false>(sC, sT, sATSw, sSwi, r);
    __syncthreads();
    // pred_inf = H*v1 + pred_conc*bu ; H[r][j] = AT_Sw[j][r]
    float pi2 = 0.0f;
    for (int j = 0; j < 32; ++j) pi2 += sATSw[j * LDP + r] * sV1[j];
    for (int j = 0; j < 32; ++j) pi2 += sC[r * LDP + j] * sBu[j];
    size_t vb = ((size_t)t * B_ + b) * D_;
    size_t cb = vb * D_;
    sInf[r] = pi2 + obs_inf[vb + r];
    // retire the async batch carrying obs_conc[t] (in-order completion):
    // 8 newer ops (for t+1) may stay outstanding.
    if (t + 1 < T_) { ASYNC_WAIT(8); } else { ASYNC_WAIT(0); }
    __syncthreads();   // pred_conc reads done before column updates below
    const float* ob = sObs[t & 1];
    for (int i = 0; i < 32; ++i) {
      float cc = sC[i * LDP + r] + ob[i * 32 + r];
      sC[i * LDP + r] = cc;
      post_conc[cb + i * 32 + r] = cc;        // coalesced
    }
    post_inf[vb + r] = sInf[r];
    __syncthreads();
  }
}

// ---------------------------------------------------------------------------
// Backward sampling + KL. One wave per (s,b) chain; post_conc is L2-resident
// (128MB < 192MB L2) and the next tile is async-prefetched into LDS.
// Uses quad(z-mu) == ||eps||^2 identity for log q.
// ---------------------------------------------------------------------------
__global__ __launch_bounds__(32) void backward_kernel(
    const float* __restrict__ u, const float* __restrict__ A,
    const float* __restrict__ Bmat,
    const float* __restrict__ prior_inf, const float* __restrict__ prior_conc,
    const float* __restrict__ eps,
    const float* __restrict__ swinv_g, const float* __restrict__ atsw_g,
    const float* __restrict__ atswa_g, const float* __restrict__ ldsw_g,
    const float* __restrict__ post_inf, const float* __restrict__ post_conc,
    float* __restrict__ out) {
  __shared__ float sM[32 * LDP], sA[32 * LDP], sSwi[32 * LDP];
  __shared__ float sATSw[32 * LDP], sATSwA[32 * LDP];
  __shared__ __align__(16) float sPc[2][D_ * D_];    // async double buffer
  __shared__ float sB[D_ * DU_];
  __shared__ float sZ[32], sV[32], sRes[32], sDv[32];
  const int r = threadIdx.x;
  const int b = blockIdx.x % B_;
  const int s = blockIdx.x / B_;
  const float ldswinv = ldsw_g[0];
  const float cst = -0.5f * (float)D_ * LOG2PI_F;

  for (int i = 0; i < 32; ++i) {
    sA[i * LDP + r]     = A[i * 32 + r];
    sSwi[i * LDP + r]   = swinv_g[i * 32 + r];
    sATSw[i * LDP + r]  = atsw_g[i * 32 + r];
    sATSwA[i * LDP + r] = atswa_g[i * 32 + r];
  }
  for (int k = r; k < D_ * DU_; k += 32) sB[k] = Bmat[k];
  __syncthreads();

  float log_q = 0.0f, log_pz = 0.0f;

  // t = T-1 : draw from last posterior (tile read synchronously),
  // and async-prefetch post_conc[T-2].
  {
    size_t vb = ((size_t)(T_ - 1) * B_ + b) * D_;
    size_t cb = vb * D_;
    sV[r] = post_inf[vb + r];
    for (int i = 0; i < 32; ++i) sM[i * LDP + r] = post_conc[cb + i * 32 + r];
    tile_load_async(post_conc + ((size_t)((T_ - 2) * B_ + b)) * D_ * D_,
                    sPc[(T_ - 2) & 1], r);
    float e = eps[(((size_t)s * T_ + (T_ - 1)) * B_ + b) * D_ + r];
    __syncthreads();
    float ld = chol32(sM, r);
    log_q += 0.5f * ld + cst - 0.5f * wredsum(e * e);
    trisolve_fwd(sM, sV, r);                   // y = L^{-1} q_inf
    sV[r] += e;                                // z = L^{-T}(y + eps)
    __syncthreads();
    trisolve_bwd(sM, sV, r);
    sZ[r] = sV[r];
    __syncthreads();
  }

  for (int t = T_ - 2; t >= 0; --t) {
    // async-prefetch the next (older) posterior tile
    if (t > 0)
      tile_load_async(post_conc + ((size_t)((t - 1) * B_ + b)) * D_ * D_,
                      sPc[(t - 1) & 1], r);
    const float* ut = u + ((size_t)t * B_ + b) * DU_;
    float bu = 0.0f;
    for (int c = 0; c < DU_; ++c) bu += sB[r * DU_ + c] * ut[c];
    sRes[r] = sZ[r] - bu;                      // resid = z_next - B u_t
    __syncthreads();
    size_t vb = ((size_t)t * B_ + b) * D_;
    float qi = post_inf[vb + r];
    for (int j = 0; j < 32; ++j) qi += sATSw[r * LDP + j] * sRes[j];
    sV[r] = qi;                                // q_inf
    // retire the async batch carrying post_conc[t] (in-order completion)
    if (t > 0) { ASYNC_WAIT(8); } else { ASYNC_WAIT(0); }
    __syncthreads();
    const float* pc = sPc[t & 1];
    for (int i = 0; i < 32; ++i)
      sM[i * LDP + r] = sATSwA[i * LDP + r] + pc[i * 32 + r];
    float e = eps[(((size_t)s * T_ + t) * B_ + b) * D_ + r];
    __syncthreads();
    float ld = chol32(sM, r);
    log_q += 0.5f * ld + cst - 0.5f * wredsum(e * e);
    trisolve_fwd(sM, sV, r);
    sV[r] += e;
    __syncthreads();
    trisolve_bwd(sM, sV, r);                   // sV = z_t
    // transition log-density: d = resid - A z_t ; quad = d^T Sw_inv d
    float az = 0.0f;
    for (int j = 0; j < 32; ++j) az += sA[r * LDP + j] * sV[j];
    float d = sRes[r] - az;
    sDv[r] = d;
    __syncthreads();
    float w = 0.0f;
    for (int j = 0; j < 32; ++j) w += sSwi[r * LDP + j] * sDv[j];
    log_pz += 0.5f * ldswinv + cst - 0.5f * wredsum(d * w);
    sZ[r] = sV[r];
    __syncthreads();
  }

  // prior log-density of z_0 (general information form)
  {
    for (int i = 0; i < 32; ++i) sM[i * LDP + r] = prior_conc[i * 32 + r];
    sV[r] = prior_inf[r];
    __syncthreads();
    float ld = chol32(sM, r);
    trisolve_fwd(sM, sV, r);
    __syncthreads();
    trisolve_bwd(sM, sV, r);                   // sV = mu0
    float d = sZ[r] - sV[r];
    sDv[r] = d;
    __syncthreads();
    // quad = ||L^T d||^2 ; (L^T d)_r = sum_{j>=r} L[j][r] d_j
    float w = 0.0f;
    for (int j = r; j < 32; ++j) w += sM[j * LDP + r] * sDv[j];
    log_pz += 0.5f * ld + cst - 0.5f * wredsum(w * w);
  }

  if (r == 0) atomicAdd(out + b, (log_q - log_pz) * (1.0f / (float)S_));
}

// ---------------------------------------------------------------------------
extern "C" void kernel_launch(void* const* d_in, const int* in_sizes, int n_in,
                              void* d_out, int out_size, void* d_ws, size_t ws_size,
                              hipStream_t stream) {
  (void)in_sizes; (void)n_in; (void)out_size; (void)ws_size;
  const float* obs_inf    = (const float*)d_in[0];
  const float* obs_conc   = (const float*)d_in[1];
  const float* u          = (const float*)d_in[2];
  const float* A          = (const float*)d_in[3];
  const float* Bmat       = (const float*)d_in[4];
  const float* Sigma_w    = (const float*)d_in[5];
  const float* prior_inf  = (const float*)d_in[6];
  const float* prior_conc = (const float*)d_in[7];
  const float* eps        = (const float*)d_in[8];
  float* out = (float*)d_out;

  float* ws        = (float*)d_ws;
  float* post_inf  = ws;                                     // T*B*D
  float* post_conc = post_inf + (size_t)T_ * B_ * D_;        // T*B*D*D (128MB, fits L2)
  float* swinv     = post_conc + (size_t)T_ * B_ * D_ * D_;  // D*D
  float* atsw      = swinv + D_ * D_;                        // D*D
  float* atswa     = atsw + D_ * D_;                         // D*D
  float* ldsw      = atswa + D_ * D_;                        // 1

  prep_kernel<<<1, 32, 0, stream>>>(A, Sigma_w, swinv, atsw, atswa, ldsw, out);
  forward_kernel<<<B_, 32, 0, stream>>>(obs_inf, obs_conc, u, Bmat,
                                        prior_inf, prior_conc,
                                        swinv, atsw, atswa, post_inf, post_conc);
  backward_kernel<<<S_ * B_, 32, 0, stream>>>(u, A, Bmat, prior_inf, prior_conc, eps,
                                              swinv, atsw, atswa, ldsw,
                                              post_inf, post_conc, out);
}